// FrequencyEnergyIndicator_32719060861553
// MI455X (gfx1250) — compile-verified
//
#include <hip/hip_runtime.h>
#include <hip/hip_bf16.h>
#include <math.h>

typedef __attribute__((ext_vector_type(2))) float v2f;
typedef __attribute__((ext_vector_type(8))) float v8f;

// Tile geometry: 64x64 output tile, halo 32 (sigma=8 -> k=65), input tile 128x128.
#define IN_STRIDE 132          // 128 + 4 pad (4-bank rotation per row)
#define H_STRIDE  68           // 64 + 4 pad
#define S_IN_ELEMS (128 * IN_STRIDE)   // 16896
#define S_H_ELEMS  (128 * H_STRIDE)    // 8704
#define S_W_ELEMS  128

__global__ __launch_bounds__(256)
void FreqEnergy_fused_kernel(const float* __restrict__ z,
                             float* __restrict__ bands,     // d_out + 48
                             float* __restrict__ partials)  // 3*16*256 floats in ws
{
    extern __shared__ float smem[];
    float* s_in = smem;                         // 128 x 132
    float* s_h  = smem + S_IN_ELEMS;            // 128 x 68  (per-sigma hres; reused as reduction buf)
    float* s_w  = smem + S_IN_ELEMS + S_H_ELEMS;// 3 packed 1-D gaussian filters

    const int tid   = threadIdx.x;
    const int lane  = tid & 31;
    const int wave  = tid >> 5;
    const int lg    = lane >> 4;     // half-wave group (selects K-pair / row group)
    const int lm    = lane & 15;

    const int plane = blockIdx.y;               // b*16 + c
    const int tile  = blockIdx.x;               // 0..15 (4x4 tiles of 64x64)
    const int ty0   = (tile >> 2) * 64;
    const int tx0   = (tile & 3) * 64;
    const float* src = z + (size_t)plane * 65536;

    // ---- filter taps: k = 17,33,65 ; halo = 8,16,32 ; sigma = 2,4,8 ----
    const int KS[3]   = {17, 33, 65};
    const int HALO[3] = {8, 16, 32};
    const int WO[3]   = {0, 20, 56};

    // unnormalized exp taps
    if (tid < 121) {
        int f = -1, tap = 0, off = 0;
        if (tid < 17)                  { f = 0; tap = tid;      off = 0;  }
        else if (tid >= 20 && tid < 53){ f = 1; tap = tid - 20; off = 20; }
        else if (tid >= 56)            { f = 2; tap = tid - 56; off = 56; }
        if (f >= 0) {
            float sig = (float)(2 << f);
            float x   = (float)tap - (float)HALO[f];
            s_w[off + tap] = expf(-(x * x) / (2.0f * sig * sig));
        }
    }

    // ---- load 128x128 zero-padded input tile (float4 fast path) ----
    for (int q = tid; q < 128 * 32; q += 256) {
        int r  = q >> 5;
        int c4 = (q & 31) << 2;
        int gy = ty0 - 32 + r;
        int gx = tx0 - 32 + c4;
        float4 v = make_float4(0.f, 0.f, 0.f, 0.f);
        if (gy >= 0 && gy < 256) {
            if ((unsigned)gx <= 252u) {
                v = *(const float4*)(src + gy * 256 + gx);
            } else {
                float* ve = (float*)&v;
                #pragma unroll
                for (int e = 0; e < 4; ++e) {
                    int gxe = gx + e;
                    ve[e] = ((unsigned)gxe < 256u) ? src[gy * 256 + gxe] : 0.f;
                }
            }
        }
        float* d = &s_in[r * IN_STRIDE + c4];
        d[0] = v.x; d[1] = v.y; d[2] = v.z; d[3] = v.w;
    }
    __syncthreads();

    // normalize each 1-D filter (g = e / sum(e); separable == reference 2-D kernel)
    if (tid < 3) {
        int off = WO[tid], k = KS[tid];
        float sum = 0.f;
        for (int i = 0; i < k; ++i) sum += s_w[off + i];
        float inv = 1.f / sum;
        for (int i = 0; i < k; ++i) s_w[off + i] *= inv;
    }
    __syncthreads();

    v8f accs[3][2];   // per-sigma vertical-pass accumulators (2 sub-tiles per wave)

    #pragma unroll
    for (int s = 0; s < 3; ++s) {
        const int   k    = KS[s];
        const int   halo = HALO[s];
        const int   K4   = (k + 15) >> 2;   // 8 / 12 / 20 WMMA steps
        const float* wp  = s_w + WO[s];

        // ---- banded-weight fragment, shared by H-pass B operand and V-pass A
        // operand: per lane, wf[t4] = { w[t0+2lg-lm], w[t0+2lg-lm+1] } (0 outside band).
        // Loads are index-clamped and UNconditional; zeroing is a pure select.
        v2f wf[20];
        #pragma unroll
        for (int t4 = 0; t4 < K4; ++t4) {
            const int d0 = (t4 << 2) + 2 * lg - lm;
            const int i0 = min(max(d0, 0), k - 1);
            const int i1 = min(max(d0 + 1, 0), k - 1);
            const float w0 = wp[i0];            // always in-bounds
            const float w1 = wp[i1];            // always in-bounds
            v2f t;
            t.x = ((unsigned)d0       < (unsigned)k) ? w0 : 0.f;
            t.y = ((unsigned)(d0 + 1) < (unsigned)k) ? w1 : 0.f;
            wf[t4] = t;
        }

        // ---- horizontal pass: hres[r, x] = sum_j w[j] * in[r, x + 32 - halo + j]
        // Out(16x16) = A_in(16xK) x Wband(Kx16)
        {
            const int ys2 = wave * 16;          // 8 waves -> rows 0..127
            for (int cb = 0; cb < 4; ++cb) {
                const int xs    = cb * 16;
                const int cbase = xs + 32 - halo;        // even
                // base index is even for every term -> 8-byte aligned v2f loads
                const v2f* arow2 = (const v2f*)&s_in[(ys2 + lm) * IN_STRIDE + cbase + 2 * lg];
                // prefetch all A fragments, then run the WMMA chain
                v2f af[20];
                #pragma unroll
                for (int t4 = 0; t4 < K4; ++t4) af[t4] = arow2[2 * t4];
                v8f acc = {};
                #pragma unroll
                for (int t4 = 0; t4 < K4; ++t4) {
                    acc = __builtin_amdgcn_wmma_f32_16x16x4_f32(
                        false, af[t4], false, wf[t4], (short)0, acc, false, false);
                }
                // D layout: VGPR i -> row (i + 8*lg), col lm
                float* hdst = &s_h[(ys2 + 8 * lg) * H_STRIDE + xs + lm];
                #pragma unroll
                for (int i = 0; i < 8; ++i) hdst[i * H_STRIDE] = acc[i];
            }
        }
        __syncthreads();

        // ---- vertical pass: out[y, x] = sum_j w[j] * hres[y + 32 - halo + j, x]
        // Out(16x16) = Wband(16xK) x B_hres(Kx16)
        #pragma unroll
        for (int i = 0; i < 2; ++i) {
            const int st = wave * 2 + i;        // 16 sub-tiles of the 64x64 output
            const int ys = (st >> 2) * 16;
            const int xs = (st & 3) * 16;
            const int rbase = ys + 32 - halo;
            const float* bcol = &s_h[(rbase + 2 * lg) * H_STRIDE + xs + lm];
            // prefetch all B fragments (rows t0, t0+1 of hres), then chain
            v2f bf[20];
            #pragma unroll
            for (int t4 = 0; t4 < K4; ++t4) {
                const float* bp = bcol + (t4 << 2) * H_STRIDE;
                v2f b; b.x = bp[0]; b.y = bp[H_STRIDE];
                bf[t4] = b;
            }
            v8f acc = {};
            #pragma unroll
            for (int t4 = 0; t4 < K4; ++t4) {
                acc = __builtin_amdgcn_wmma_f32_16x16x4_f32(
                    false, wf[t4], false, bf[t4], (short)0, acc, false, false);
            }
            accs[s][i] = acc;
        }
        __syncthreads();   // s_h will be overwritten next sigma (or reused for reduction)
    }

    // ---- bands + energy ----
    // band0 = blur8 ; band1 = blur4 - blur8 ; band2 = blur2 - blur4
    float e0 = 0.f, e1 = 0.f, e2 = 0.f;
    #pragma unroll
    for (int i = 0; i < 2; ++i) {
        const int st = wave * 2 + i;
        const int ys = (st >> 2) * 16;
        const int xs = (st & 3) * 16;
        const int col = tx0 + xs + lm;
        #pragma unroll
        for (int j = 0; j < 8; ++j) {
            const int row = ty0 + ys + j + 8 * lg;
            const size_t off = (size_t)row * 256 + col;
            const float v2v = accs[0][i][j];
            const float v4v = accs[1][i][j];
            const float v8v = accs[2][i][j];
            const float b0 = v8v;
            const float b1 = v4v - v8v;
            const float b2 = v2v - v4v;
            bands[((size_t)(0 * 256 + plane)) * 65536 + off] = b0;
            bands[((size_t)(1 * 256 + plane)) * 65536 + off] = b1;
            bands[((size_t)(2 * 256 + plane)) * 65536 + off] = b2;
            e0 += b0 * b0; e1 += b1 * b1; e2 += b2 * b2;
        }
    }

    // ---- deterministic block reduction of energies (reuse s_h) ----
    float* red = s_h;
    red[tid]       = e0;
    red[256 + tid] = e1;
    red[512 + tid] = e2;
    __syncthreads();
    for (int o = 128; o > 0; o >>= 1) {
        if (tid < o) {
            red[tid]       += red[tid + o];
            red[256 + tid] += red[256 + tid + o];
            red[512 + tid] += red[512 + tid + o];
        }
        __syncthreads();
    }
    if (tid == 0) {
        const int b = plane >> 4, c = plane & 15;
        const int slot = c * 16 + tile;           // 256 deterministic slots per (s,b)
        partials[(0 * 16 + b) * 256 + slot] = red[0];
        partials[(1 * 16 + b) * 256 + slot] = red[256];
        partials[(2 * 16 + b) * 256 + slot] = red[512];
    }
}

__global__ void FreqEnergy_finalize_kernel(const float* __restrict__ partials,
                                           float* __restrict__ out)
{
    __shared__ float E[48];
    const int t = threadIdx.x;
    if (t < 48) {
        const float* p = partials + t * 256;
        float sum = 0.f;
        for (int i = 0; i < 256; ++i) sum += p[i];
        E[t] = sum;                                // E[s*16 + b]
    }
    __syncthreads();
    if (t < 16) {
        const float es0 = E[0 * 16 + t];
        const float es1 = E[1 * 16 + t];
        const float es2 = E[2 * 16 + t];
        const float inv = 1.f / (es0 + es1 + es2 + 1e-8f);
        out[t * 3 + 0] = es0 * inv;                // e_t is (16, 3) row-major
        out[t * 3 + 1] = es1 * inv;
        out[t * 3 + 2] = es2 * inv;
    }
}

extern "C" void kernel_launch(void* const* d_in, const int* in_sizes, int n_in,
                              void* d_out, int out_size, void* d_ws, size_t ws_size,
                              hipStream_t stream) {
    const float* z = (const float*)d_in[0];
    float* out = (float*)d_out;
    float* partials = (float*)d_ws;                // 3*16*256 floats = 48 KB

    dim3 grid(16, 256);                            // 4x4 tiles  x  (B*C = 256) planes
    dim3 block(256);
    size_t shmem = (size_t)(S_IN_ELEMS + S_H_ELEMS + S_W_ELEMS) * sizeof(float);

    FreqEnergy_fused_kernel<<<grid, block, shmem, stream>>>(z, out + 48, partials);
    FreqEnergy_finalize_kernel<<<1, 64, 0, stream>>>(partials, out);
}